// Prototype_83897891160157
// MI455X (gfx1250) — compile-verified
//
#include <hip/hip_runtime.h>

#define N_ROWS   1000000
#define FEAT     256
#define NCLASS   345
#define NDOM     6
#define NSEG     (NCLASS * NDOM)     // 2070
#define BETA     0.5f
#define EPSF     1e-5f

#define SLICE    16                  // feature columns per accumulate block
#define NSLICES  (FEAT / SLICE)      // 16
#define NCHUNK   64                  // row chunks
#define ROWS_PER_CHUNK ((N_ROWS + NCHUNK - 1) / NCHUNK)   // 15625

typedef float v2f __attribute__((ext_vector_type(2)));
typedef float v8f __attribute__((ext_vector_type(8)));

// ---------------------------------------------------------------------------
// Kernel A: zero the accumulator region (d_out doubles as sums||counts).
// ---------------------------------------------------------------------------
__global__ void zero_kernel(float* __restrict__ p, int n) {
    int i = blockIdx.x * blockDim.x + threadIdx.x;
    if (i < n) p[i] = 0.0f;
}

// ---------------------------------------------------------------------------
// Kernel B: scatter-reduce with full LDS privatization.
// Each block owns one 16-feature slice; LDS holds all 2070 segments for it.
// ---------------------------------------------------------------------------
__global__ __launch_bounds__(256) void accum_kernel(
    const float* __restrict__ feats,
    const int*   __restrict__ labels,
    const int*   __restrict__ domains,
    float*       __restrict__ gsums,     // [NSEG * FEAT]
    float*       __restrict__ gcounts)   // [NSEG]
{
    __shared__ float lsums[NSEG * SLICE];   // 132480 B
    __shared__ float lcnt[NSEG];            //   8280 B

    const int slice = blockIdx.x & (NSLICES - 1);
    const int chunk = blockIdx.x / NSLICES;

    for (int i = threadIdx.x; i < NSEG * SLICE; i += 256) lsums[i] = 0.0f;
    for (int i = threadIdx.x; i < NSEG; i += 256) lcnt[i] = 0.0f;
    __syncthreads();

    const int q  = threadIdx.x & 3;      // which float4 of the slice
    const int lr = threadIdx.x >> 2;     // row lane 0..63
    const int fbase = slice * SLICE + q * 4;

    int row0 = chunk * ROWS_PER_CHUNK;
    int row1 = row0 + ROWS_PER_CHUNK;
    if (row1 > N_ROWS) row1 = N_ROWS;
    const bool do_count = (slice == 0) && (q == 0);

    for (int r = row0 + lr; r < row1; r += 64) {
        // speculative prefetch of next iteration's 64B -> global_prefetch_b8
        __builtin_prefetch(feats + (size_t)(r + 64) * FEAT + fbase, 0, 1);

        const int seg = domains[r] * NCLASS + labels[r];
        const float4 v = *reinterpret_cast<const float4*>(
            feats + (size_t)r * FEAT + fbase);
        float* dst = &lsums[seg * SLICE + (q << 2)];
        atomicAdd(dst + 0, v.x);             // ds_add_f32
        atomicAdd(dst + 1, v.y);
        atomicAdd(dst + 2, v.z);
        atomicAdd(dst + 3, v.w);
        if (do_count) atomicAdd(&lcnt[seg], 1.0f);
    }
    __syncthreads();

    // Flush privatized accumulators to global (L2-resident, 2.1 MB)
    for (int i = threadIdx.x; i < NSEG * SLICE; i += 256) {
        const float v = lsums[i];
        if (v != 0.0f) {
            const int seg = i >> 4;          // i / SLICE
            const int fl  = i & (SLICE - 1);
            atomicAdd(&gsums[(size_t)seg * FEAT + slice * SLICE + fl], v);
        }
    }
    if (slice == 0) {
        for (int i = threadIdx.x; i < NSEG; i += 256) {
            const float v = lcnt[i];
            if (v != 0.0f) atomicAdd(&gcounts[i], v);
        }
    }
}

// ---------------------------------------------------------------------------
// Kernel C: finalize in place. One wave per 16 segments. The 256-wide
// row-sum (for the mask) is done with chained V_WMMA_F32_16X16X4_F32 against
// an all-ones B: D[m][n] = sum_k A[m][k], accumulated over 64 K-chunks.
// ---------------------------------------------------------------------------
__global__ __launch_bounds__(256) void finalize_kernel(
    const float* __restrict__ mean_in,    // [NSEG * FEAT]
    const float* __restrict__ amount_in,  // [NSEG]
    float*       __restrict__ out)        // sums||counts -> new_mean||new_amount
{
    float* gsums   = out;
    float* gamount = out + (size_t)NSEG * FEAT;

    const int wave = threadIdx.x >> 5;
    const int lane = threadIdx.x & 31;
    const int group = blockIdx.x * 8 + wave;
    const int segbase = group * 16;
    if (segbase >= NSEG) return;          // wave-uniform exit, EXEC stays full

    // A-matrix (16x4 f32): lanes 0-15 hold K=0,1 for M=lane;
    // lanes 16-31 hold K=2,3 for M=lane-16. Clamp seg (branchless) so the
    // last partial tile keeps EXEC all-ones through the WMMAs.
    const int myseg0 = segbase + (lane & 15);
    const int myseg  = (myseg0 < NSEG) ? myseg0 : (NSEG - 1);
    const float inv  = 1.0f / (gamount[myseg] + EPSF);
    const float* rowp = gsums + (size_t)myseg * FEAT + ((lane >> 4) << 1);

    v2f b; b.x = 1.0f; b.y = 1.0f;        // ones => any B layout gives rowsum
    v8f c = {};
    for (int kc = 0; kc < FEAT / 4; ++kc) {
        v2f a;
        a.x = rowp[kc * 4 + 0] * inv;     // tmp_mean element
        a.y = rowp[kc * 4 + 1] * inv;
        c = __builtin_amdgcn_wmma_f32_16x16x4_f32(
                false, a, false, b, (short)0, c, false, false);
    }

    // C/D layout: VGPR v, lane 0 -> (M=v, N=0); lane 16 -> (M=v+8, N=0)
#pragma unroll
    for (int m = 0; m < 16; ++m) {
        const int seg = segbase + m;
        if (seg >= NSEG) break;
        const int rsbits = (m < 8)
            ? __builtin_amdgcn_readlane(__builtin_bit_cast(int, c[m]), 0)
            : __builtin_amdgcn_readlane(__builtin_bit_cast(int, c[m - 8]), 16);
        const float rs  = __builtin_bit_cast(float, rsbits);
        const float msk = (rs != 0.0f) ? 1.0f : 0.0f;

        const float cm   = gamount[seg];
        const float invm = 1.0f / (cm + EPSF);
        const size_t base = (size_t)seg * FEAT;
        for (int f = lane; f < FEAT; f += 32) {
            const float tm = gsums[base + f] * invm;
            const float mn = mean_in[base + f];
            const float blended = mn * BETA + tm * (1.0f - BETA);
            gsums[base + f] = mn * (1.0f - msk) + blended * msk;  // new_mean
        }
        if (lane == 0)
            gamount[seg] = (cm > 0.0f) ? 1.0f : amount_in[seg];   // new_amount
    }
}

// ---------------------------------------------------------------------------
extern "C" void kernel_launch(void* const* d_in, const int* in_sizes, int n_in,
                              void* d_out, int out_size, void* d_ws, size_t ws_size,
                              hipStream_t stream) {
    const float* feats   = (const float*)d_in[0];
    const float* mean    = (const float*)d_in[1];
    const float* amount  = (const float*)d_in[2];
    const int*   labels  = (const int*)d_in[3];
    const int*   domains = (const int*)d_in[4];
    float* out = (float*)d_out;   // [NSEG*FEAT new_mean][NSEG new_amount]

    const int ztotal = NSEG * FEAT + NSEG;   // 531990 == out_size
    zero_kernel<<<dim3((ztotal + 255) / 256), dim3(256), 0, stream>>>(out, ztotal);

    accum_kernel<<<dim3(NSLICES * NCHUNK), dim3(256), 0, stream>>>(
        feats, labels, domains, out, out + (size_t)NSEG * FEAT);

    // 130 groups of 16 segments, 8 waves (groups) per block -> 17 blocks
    finalize_kernel<<<dim3(17), dim3(256), 0, stream>>>(mean, amount, out);
}